// DsvtToDenseBEV_89721866813443
// MI455X (gfx1250) — compile-verified
//
#include <hip/hip_runtime.h>
#include <stdint.h>

// DSVT -> dense BEV scatter-max for MI455X (gfx1250).
//
// Pure data movement (no matmul structure -> WMMA inapplicable). CDNA5-specific
// paths used instead:
//   * global_atomic_max_num_f32 (inline asm, device scope) — IEEE maximumNumber,
//     NaN-in-memory means "absent", which kills the usual decode/fixup pass.
//   * wave32 ballot/popc prefix ranking to compact touched cells into an
//     L2-resident (<=76.8MB < 192MB) accumulator so atomics never thrash HBM.
//   * th:TH_NT (via __builtin_nontemporal_*) on the read-once feature stream
//     and the write-once 327MB output so they don't evict the compact buffer.
// Roofline: ~0.5 GB moved @ 23.3 TB/s ≈ 21 us (floor ≈ 17 us).

namespace {

constexpr uint32_t NXC = 400u;
constexpr uint32_t NYC = 400u;
constexpr uint32_t CC  = 128u;
constexpr uint32_t NAN_BITS = 0xFFFFFFFFu; // quiet NaN sentinel (all ones)

typedef float    f32x4 __attribute__((ext_vector_type(4)));
typedef uint32_t u32x4 __attribute__((ext_vector_type(4)));

__device__ __forceinline__ void global_atomic_fmax(float* p, float v) {
  // Non-returning device-scope float max atomic (tracked with STOREcnt).
  asm volatile("global_atomic_max_num_f32 %0, %1, off scope:SCOPE_DEV"
               :
               : "v"(p), "v"(v)
               : "memory");
}

// ---------------- shared helpers ----------------

__global__ void k_zero_u32(uint32_t* __restrict__ p, uint32_t n) {
  uint32_t i = blockIdx.x * blockDim.x + threadIdx.x;
  if (i < n) p[i] = 0u;
}

// RT fill: the compact accumulator should become/stay L2-resident.
__global__ void k_fill_u32x4(uint32_t* __restrict__ p, uint32_t n4, uint32_t val) {
  uint32_t i = blockIdx.x * blockDim.x + threadIdx.x;
  if (i < n4) {
    u32x4 v = {val, val, val, val};
    reinterpret_cast<u32x4*>(p)[i] = v;
  }
}

// Mark touched cells in a bitmap (1 bit per BEV cell).
__global__ void k_mark(const int* __restrict__ bidx, const int* __restrict__ yidx,
                       const int* __restrict__ xidx, uint32_t* __restrict__ bitmap,
                       uint32_t nvox) {
  uint32_t i = blockIdx.x * blockDim.x + threadIdx.x;
  if (i >= nvox) return;
  uint32_t cell = (uint32_t)bidx[i] * (NYC * NXC) +
                  (uint32_t)yidx[i] * NXC + (uint32_t)xidx[i];
  atomicOr(&bitmap[cell >> 5], 1u << (cell & 31u));
}

// ---------------- compact (L2-resident) path ----------------

// Per-256-cell block: count touched cells (wave32 ballot + popc).
__global__ void k_blocksum(const uint32_t* __restrict__ bitmap,
                           uint32_t* __restrict__ bsum, uint32_t ncells) {
  uint32_t cell = blockIdx.x * 256u + threadIdx.x;
  bool flag = (cell < ncells) ? (((bitmap[cell >> 5] >> (cell & 31u)) & 1u) != 0u) : false;
  uint32_t bal = __builtin_amdgcn_ballot_w32(flag);
  __shared__ uint32_t wsum[8];
  uint32_t lane = threadIdx.x & 31u, wave = threadIdx.x >> 5;
  if (lane == 0u) wsum[wave] = (uint32_t)__popc(bal);
  __syncthreads();
  if (threadIdx.x == 0u) {
    uint32_t s = 0u;
    for (int w = 0; w < 8; ++w) s += wsum[w];
    bsum[blockIdx.x] = s;
  }
}

// Single-block exclusive scan over the (2500) block sums.
__global__ void k_scan_exclusive(uint32_t* __restrict__ a, uint32_t n) {
  __shared__ uint32_t sh[256];
  __shared__ uint32_t carry;
  if (threadIdx.x == 0u) carry = 0u;
  __syncthreads();
  for (uint32_t base = 0u; base < n; base += 256u) {
    uint32_t i = base + threadIdx.x;
    uint32_t v = (i < n) ? a[i] : 0u;
    sh[threadIdx.x] = v;
    __syncthreads();
    for (uint32_t off = 1u; off < 256u; off <<= 1u) {
      uint32_t t = (threadIdx.x >= off) ? sh[threadIdx.x - off] : 0u;
      __syncthreads();
      sh[threadIdx.x] += t;
      __syncthreads();
    }
    uint32_t excl = sh[threadIdx.x] - v + carry;   // reads carry before update below
    if (i < n) a[i] = excl;
    uint32_t tot = sh[255];
    __syncthreads();
    if (threadIdx.x == 0u) carry += tot;
    __syncthreads();
  }
}

// rank[cell] = exclusive count of touched cells before `cell`.
__global__ void k_rank(const uint32_t* __restrict__ bitmap,
                       const uint32_t* __restrict__ bsum,
                       uint32_t* __restrict__ rank, uint32_t ncells) {
  uint32_t cell = blockIdx.x * 256u + threadIdx.x;
  bool flag = (cell < ncells) ? (((bitmap[cell >> 5] >> (cell & 31u)) & 1u) != 0u) : false;
  uint32_t bal = __builtin_amdgcn_ballot_w32(flag);
  uint32_t lane = threadIdx.x & 31u, wave = threadIdx.x >> 5;
  uint32_t laneExcl = (uint32_t)__popc(bal & ((1u << lane) - 1u));
  __shared__ uint32_t wsum[8];
  if (lane == 0u) wsum[wave] = (uint32_t)__popc(bal);
  __syncthreads();
  uint32_t waveOff = 0u;
  for (uint32_t w = 0u; w < wave; ++w) waveOff += wsum[w];
  if (cell < ncells) rank[cell] = bsum[blockIdx.x] + waveOff + laneExcl;
}

// 2 voxels per 256-thread block; lane c handles channel c. Feature loads are
// fully coalesced (512B per voxel) and non-temporal (read exactly once) so
// they don't evict the compact accumulator from L2. Atomics land in the
// compact buffer, which fits in L2, instead of the 327MB output.
__global__ void k_scatter_compact(const float* __restrict__ feat,
                                  const int* __restrict__ bidx,
                                  const int* __restrict__ yidx,
                                  const int* __restrict__ xidx,
                                  const uint32_t* __restrict__ rank,
                                  float* __restrict__ compact,
                                  uint32_t slotStride, uint32_t nvox) {
  uint32_t voxel = blockIdx.x * 2u + (threadIdx.x >> 7);
  uint32_t c = threadIdx.x & 127u;
  if (voxel >= nvox) return;
  uint32_t cell = (uint32_t)bidx[voxel] * (NYC * NXC) +
                  (uint32_t)yidx[voxel] * NXC + (uint32_t)xidx[voxel];
  uint32_t slot = rank[cell];
  float v = __builtin_nontemporal_load(&feat[(size_t)voxel * CC + c]);
  global_atomic_fmax(&compact[(size_t)c * slotStride + slot], v);
}

// One float4 (4 consecutive x) per thread: the only pass that touches the
// full output, written exactly once, coalesced, non-temporal (never re-read;
// keeps the compact buffer resident while we drain it).
__global__ void k_final_compact(const uint32_t* __restrict__ bitmap,
                                const uint32_t* __restrict__ rank,
                                const float* __restrict__ compact,
                                float* __restrict__ out,
                                uint32_t slotStride, uint32_t nquads) {
  uint32_t q = blockIdx.x * blockDim.x + threadIdx.x;
  if (q >= nquads) return;
  uint32_t flat = q * 4u;               // ((b*C + c)*NY + y)*NX + x, x % 4 == 0
  uint32_t x  = flat % NXC;
  uint32_t t  = flat / NXC;
  uint32_t y  = t % NYC;
  uint32_t bc = t / NYC;                // b*C + c
  uint32_t c  = bc % CC;
  uint32_t b  = bc / CC;
  uint32_t cell0 = (b * NYC + y) * NXC + x;     // multiple of 4 -> 4 bits in one word
  uint32_t bits = (bitmap[cell0 >> 5] >> (cell0 & 31u)) & 0xFu;
  u32x4 r = reinterpret_cast<const u32x4*>(rank)[cell0 >> 2];
  f32x4 o;
  o.x = (bits & 1u) ? compact[(size_t)c * slotStride + r.x] : 0.0f;
  o.y = (bits & 2u) ? compact[(size_t)c * slotStride + r.y] : 0.0f;
  o.z = (bits & 4u) ? compact[(size_t)c * slotStride + r.z] : 0.0f;
  o.w = (bits & 8u) ? compact[(size_t)c * slotStride + r.w] : 0.0f;
  __builtin_nontemporal_store(o, reinterpret_cast<f32x4*>(out) + q);
}

// ---------------- fallback path (tiny workspace): atomics direct into out ----------------

// Prefill out: touched cells -> NaN sentinel (first atomic installs value),
// untouched -> 0 (final value). No fixup pass needed afterwards.
__global__ void k_prefill_out(const uint32_t* __restrict__ bitmap,
                              uint32_t* __restrict__ out_u32, uint32_t nquads) {
  uint32_t q = blockIdx.x * blockDim.x + threadIdx.x;
  if (q >= nquads) return;
  uint32_t flat = q * 4u;
  uint32_t x  = flat % NXC;
  uint32_t t  = flat / NXC;
  uint32_t y  = t % NYC;
  uint32_t bc = t / NYC;
  uint32_t b  = bc / CC;
  uint32_t cell0 = (b * NYC + y) * NXC + x;
  uint32_t bits = (bitmap[cell0 >> 5] >> (cell0 & 31u)) & 0xFu;
  u32x4 o;
  o.x = (bits & 1u) ? NAN_BITS : 0u;
  o.y = (bits & 2u) ? NAN_BITS : 0u;
  o.z = (bits & 4u) ? NAN_BITS : 0u;
  o.w = (bits & 8u) ? NAN_BITS : 0u;
  reinterpret_cast<u32x4*>(out_u32)[q] = o;
}

__global__ void k_scatter_direct(const float* __restrict__ feat,
                                 const int* __restrict__ bidx,
                                 const int* __restrict__ yidx,
                                 const int* __restrict__ xidx,
                                 float* __restrict__ out, uint32_t nvox) {
  uint32_t voxel = blockIdx.x * 2u + (threadIdx.x >> 7);
  uint32_t c = threadIdx.x & 127u;
  if (voxel >= nvox) return;
  uint32_t b = (uint32_t)bidx[voxel];
  uint32_t y = (uint32_t)yidx[voxel];
  uint32_t x = (uint32_t)xidx[voxel];
  float v = __builtin_nontemporal_load(&feat[(size_t)voxel * CC + c]);
  size_t o = (((size_t)(b * CC + c) * NYC) + y) * NXC + x;
  global_atomic_fmax(&out[o], v);
}

inline size_t align256(size_t v) { return (v + 255u) & ~(size_t)255u; }

} // namespace

extern "C" void kernel_launch(void* const* d_in, const int* in_sizes, int n_in,
                              void* d_out, int out_size, void* d_ws, size_t ws_size,
                              hipStream_t stream) {
  const float* feat = (const float*)d_in[0];
  const int* bidx   = (const int*)d_in[1];
  const int* yidx   = (const int*)d_in[2];
  const int* xidx   = (const int*)d_in[3];
  (void)n_in;

  const uint32_t nvox   = (uint32_t)in_sizes[1];               // voxel count
  const uint32_t B      = (uint32_t)out_size / (CC * NYC * NXC);
  const uint32_t ncells = B * NYC * NXC;                        // 640000
  const uint32_t nquads = (uint32_t)out_size / 4u;              // out_size % 4 == 0

  const uint32_t bitmapWords = (ncells + 31u) / 32u;
  const uint32_t nScanBlocks = (ncells + 255u) / 256u;          // 2500

  // Workspace layout (compact path)
  const size_t off_bsum    = align256((size_t)bitmapWords * 4u);
  const size_t off_rank    = align256(off_bsum + (size_t)nScanBlocks * 4u);
  const size_t off_compact = align256(off_rank + (size_t)ncells * 4u);
  const uint32_t maxTouched = (nvox < ncells) ? nvox : ncells;
  const uint32_t slotStride = ((maxTouched + 127u) / 128u) * 128u; // elems, non-pow2 stride
  const size_t compactBytes = (size_t)slotStride * CC * 4u;
  const bool compactPath = ws_size >= off_compact + compactBytes;

  uint32_t* bitmap = (uint32_t*)d_ws;

  // 1) zero bitmap, 2) mark touched cells (shared by both paths)
  k_zero_u32<<<(bitmapWords + 255u) / 256u, 256, 0, stream>>>(bitmap, bitmapWords);
  k_mark<<<(nvox + 255u) / 256u, 256, 0, stream>>>(bidx, yidx, xidx, bitmap, nvox);

  if (compactPath) {
    uint32_t* bsum    = (uint32_t*)((char*)d_ws + off_bsum);
    uint32_t* rank    = (uint32_t*)((char*)d_ws + off_rank);
    float*    compact = (float*)((char*)d_ws + off_compact);

    k_blocksum<<<nScanBlocks, 256, 0, stream>>>(bitmap, bsum, ncells);
    k_scan_exclusive<<<1, 256, 0, stream>>>(bsum, nScanBlocks);
    k_rank<<<nScanBlocks, 256, 0, stream>>>(bitmap, bsum, rank, ncells);

    const uint32_t n4 = (slotStride * CC) / 4u;                 // slotStride % 128 == 0
    k_fill_u32x4<<<(n4 + 255u) / 256u, 256, 0, stream>>>((uint32_t*)compact, n4, NAN_BITS);

    k_scatter_compact<<<(nvox + 1u) / 2u, 256, 0, stream>>>(
        feat, bidx, yidx, xidx, rank, compact, slotStride, nvox);

    k_final_compact<<<(nquads + 255u) / 256u, 256, 0, stream>>>(
        bitmap, rank, compact, (float*)d_out, slotStride, nquads);
  } else {
    // Tiny-workspace fallback: NaN/0 prefill + device-scope fmax atomics into out.
    k_prefill_out<<<(nquads + 255u) / 256u, 256, 0, stream>>>(
        bitmap, (uint32_t*)d_out, nquads);
    k_scatter_direct<<<(nvox + 1u) / 2u, 256, 0, stream>>>(
        feat, bidx, yidx, xidx, (float*)d_out, nvox);
  }
}